// MPNN_CONV_24850680775472
// MI455X (gfx1250) — compile-verified
//
#include <hip/hip_runtime.h>

// Problem constants from the reference
#define BATCH 4096
#define NODES 64
#define FDIM  128
#define LDS_S 132   // padded dword stride: 16B-aligned rows (132%4==0), 132%64==4
                    // => A-fragment ds_load_b64 banks = 4m + const : conflict-free

typedef __attribute__((ext_vector_type(2))) float v2f;
typedef __attribute__((ext_vector_type(8))) float v8f;

// ---------------------------------------------------------------------------
// Kernel 1: xbar[b, f] = mean over c of emb[b, c, f]
// One wave32 per batch row; lane owns one float4 (4 of 128 floats) of the row.
// Each c-iteration the wave issues a fully coalesced 512B global_load_b128.
// 128 MiB read total -> this kernel IS the roofline (~5.6us at 23.3 TB/s).
// ---------------------------------------------------------------------------
__launch_bounds__(256)
__global__ void gcn_mean_kernel(const float* __restrict__ emb,
                                float* __restrict__ xbar) {
    const int wave = threadIdx.x >> 5;          // 0..7
    const int lane = threadIdx.x & 31;
    const int b = blockIdx.x * 8 + wave;        // grid = 512 blocks

    const float4* __restrict__ src =
        (const float4*)(emb + (size_t)b * (NODES * FDIM));

    float4 acc = make_float4(0.f, 0.f, 0.f, 0.f);
#pragma unroll 16
    for (int c = 0; c < NODES; ++c) {
        float4 v = src[c * (FDIM / 4) + lane];
        acc.x += v.x; acc.y += v.y; acc.z += v.z; acc.w += v.w;
    }
    const float s = 1.0f / (float)NODES;
    acc.x *= s; acc.y *= s; acc.z *= s; acc.w *= s;
    ((float4*)(xbar + (size_t)b * FDIM))[lane] = acc;
}

// ---------------------------------------------------------------------------
// Kernel 2: fused 4-stage chain on [4096,128] vectors:
//   y1 = relu(xbar@W0+b0); y2 = relu(y1@W1+b1); y3 = relu(y2@W2+b2)
//   out = y3@Wr + br
// Block = 256 threads (8 waves) owns 16 batch rows.
// Wave w computes output columns [16w, 16w+16) of the 16x128 slab using
// v_wmma_f32_16x16x4_f32 with K accumulated over 32 steps.
// ---------------------------------------------------------------------------
__launch_bounds__(256)
__global__ void gcn_mlp_kernel(const float* __restrict__ xbar,
                               const float* __restrict__ W0, const float* __restrict__ b0,
                               const float* __restrict__ W1, const float* __restrict__ b1,
                               const float* __restrict__ W2, const float* __restrict__ b2,
                               const float* __restrict__ Wr, const float* __restrict__ br,
                               float* __restrict__ out) {
    __shared__ float Wlds[FDIM * LDS_S];        // 128x128 weights, padded rows
    __shared__ float ybuf[2][16 * LDS_S];       // activation slab ping/pong

    const int tid   = threadIdx.x;
    const int wave  = tid >> 5;                 // 0..7 -> N tile
    const int lane  = tid & 31;
    const int b0row = blockIdx.x * 16;          // grid = 256 blocks

    // Stage y0 = xbar rows [b0row, b0row+16) into ybuf[0] (2048 floats).
#pragma unroll
    for (int i = 0; i < 2; ++i) {
        int fi = tid + i * 256;                 // float4 id 0..511
        int m  = fi >> 5;                       // row in slab
        int f  = (fi & 31) << 2;                // column
        float4 v = ((const float4*)(xbar + (size_t)(b0row + m) * FDIM))[fi & 31];
        *(float4*)&ybuf[0][m * LDS_S + f] = v;
    }

    const float* Ws[4] = {W0, W1, W2, Wr};
    const float* bs[4] = {b0, b1, b2, br};

    const int m     = lane & 15;                // output row within tile
    const int khalf = lane >> 4;                // 0|1 : which K-pair this lane holds
    const int n     = (lane & 15) + wave * 16;  // output column (0..127)

    int cur = 0;
#pragma unroll
    for (int layer = 0; layer < 4; ++layer) {
        // Barrier: prior iteration's yout writes + Wlds reads are complete.
        __syncthreads();

        // Cooperative stage of this layer's 128x128 weights (4096 float4s).
        const float* __restrict__ W = Ws[layer];
#pragma unroll 4
        for (int i = 0; i < 16; ++i) {
            int fi = i * 256 + tid;             // float4 id 0..4095
            int k  = fi >> 5;                   // weight row (input dim)
            int nn = (fi & 31) << 2;            // weight col (output dim)
            float4 v = ((const float4*)(W + (size_t)k * FDIM))[fi & 31];
            *(float4*)&Wlds[k * LDS_S + nn] = v;
        }
        const float bias = bs[layer][n];        // one scalar per lane
        __syncthreads();

        // 16x16 f32 tile accumulate: D += A(16x4) x B(4x16), K = 128.
        const float* __restrict__ yin = ybuf[cur];
        v8f acc = {0.f, 0.f, 0.f, 0.f, 0.f, 0.f, 0.f, 0.f};
#pragma unroll 4
        for (int t = 0; t < 32; ++t) {
            const int kb = t * 4;
            // A fragment: lane holds A[m, kb+2*khalf .. +1]  (8B-aligned ds_load_b64)
            v2f a = *(const v2f*)&yin[m * LDS_S + kb + 2 * khalf];
            // B fragment: lane holds B[kb+2*khalf, n] and B[kb+2*khalf+1, n]
            v2f bf;
            bf.x = Wlds[(kb + 2 * khalf + 0) * LDS_S + n];
            bf.y = Wlds[(kb + 2 * khalf + 1) * LDS_S + n];
            acc = __builtin_amdgcn_wmma_f32_16x16x4_f32(
                false, a, false, bf, (short)0, acc, false, false);
        }

        // C/D layout: acc[r] of this lane is element (M = r + 8*khalf, N = n).
        if (layer < 3) {
            float* __restrict__ yout = ybuf[1 - cur];
#pragma unroll
            for (int r = 0; r < 8; ++r) {
                float v = acc[r] + bias;
                yout[(r + khalf * 8) * LDS_S + n] = v > 0.f ? v : 0.f;  // ReLU
            }
            cur = 1 - cur;
        } else {
#pragma unroll
            for (int r = 0; r < 8; ++r) {
                out[(size_t)(b0row + r + khalf * 8) * FDIM + n] = acc[r] + bias;
            }
        }
    }
}

// ---------------------------------------------------------------------------
// d_in order (setup_inputs): 0=embeddings, 1=W0, 2=b0, 3=W1, 4=b1,
//                            5=W2, 6=b2, 7=Wr, 8=br. All float32.
// d_out: [4096,128] float32. d_ws: xbar scratch (2 MiB needed).
// ---------------------------------------------------------------------------
extern "C" void kernel_launch(void* const* d_in, const int* in_sizes, int n_in,
                              void* d_out, int out_size, void* d_ws, size_t ws_size,
                              hipStream_t stream) {
    const float* emb = (const float*)d_in[0];
    const float* W0  = (const float*)d_in[1];
    const float* b0  = (const float*)d_in[2];
    const float* W1  = (const float*)d_in[3];
    const float* b1  = (const float*)d_in[4];
    const float* W2  = (const float*)d_in[5];
    const float* b2  = (const float*)d_in[6];
    const float* Wr  = (const float*)d_in[7];
    const float* br  = (const float*)d_in[8];
    float* out  = (float*)d_out;
    float* xbar = (float*)d_ws;   // BATCH*FDIM floats = 2 MiB

    gcn_mean_kernel<<<BATCH / 8, 256, 0, stream>>>(emb, xbar);
    gcn_mlp_kernel<<<BATCH / 16, 256, 0, stream>>>(
        xbar, W0, b0, W1, b1, W2, b2, Wr, br, out);
}